// CmapEncdoer_20263655702714
// MI455X (gfx1250) — compile-verified
//
#include <hip/hip_runtime.h>
#include <hip/hip_bf16.h>

typedef __attribute__((ext_vector_type(2))) float v2f;
typedef __attribute__((ext_vector_type(8))) float v8f;

#define N_NODES 50000
#define IN_CH   256
#define OUT_CH  200

// LDS slab: per matrix, 16 columns x 256 K values, K-major with stride 260
// (260 % 64 == 4 -> ds_load_b64 across the wave touches banks 4n+j, all 64
// banks exactly once: conflict-free).
#define LW_STRIDE 260
#define LW_MAT    (16 * LW_STRIDE)

// ---------------------------------------------------------------------------
// 1) degree init: every node starts at 1.0 (its self-loop)
__global__ void deg_init_kernel(float* __restrict__ deg) {
    int i = blockIdx.x * blockDim.x + threadIdx.x;
    if (i < N_NODES) deg[i] = 1.0f;
}

// 2) accumulate in-degree over real edges
__global__ void deg_accum_kernel(const long long* __restrict__ dst,
                                 float* __restrict__ deg, int E) {
    int e = blockIdx.x * blockDim.x + threadIdx.x;
    if (e < E) atomicAdd(&deg[(int)dst[e]], 1.0f);
}

// 3) dinv = rsqrt(deg) in place
__global__ void deg_rsqrt_kernel(float* __restrict__ deg) {
    int i = blockIdx.x * blockDim.x + threadIdx.x;
    if (i < N_NODES) {
        float d = deg[i];
        deg[i] = (d > 0.0f) ? rsqrtf(d) : 0.0f;
    }
}

// ---------------------------------------------------------------------------
// 4) fused dual GEMM: xw1 = x@W1, xw2 = x@W2 using V_WMMA_F32_16X16X4_F32.
// Block = 160 threads = 5 waves -> 5 consecutive row tiles, one col tile.
// W column slabs staged in LDS (zero-padded past col 200), K-major transposed
// so each lane's B fragment is one 8B-aligned ds_load_b64, bank-conflict-free.
// grid = (13 col tiles, 625 row groups); 625*5*16 = 50000 rows exactly.
__global__ __launch_bounds__(160)
void gcn_gemm_wmma(const float* __restrict__ x,
                   const float* __restrict__ W1,
                   const float* __restrict__ W2,
                   float* __restrict__ xw1,
                   float* __restrict__ xw2) {
    __shared__ float lw[2 * LW_MAT];   // ~33 KB of the 320 KB WGP pool

    const int tid  = threadIdx.x;
    const int wave = tid >> 5;
    const int lane = tid & 31;
    const int rowBase = (blockIdx.y * 5 + wave) * 16;
    const int colBase = blockIdx.x * 16;

    // Cooperative slab fill: i -> (k = i/16, n_local = i%16) so global reads
    // are 16-wide coalesced; LDS writes hit banks (4*n_local + k) % 64.
    for (int i = tid; i < 16 * IN_CH; i += 160) {
        int k  = i >> 4;
        int nl = i & 15;
        int n  = colBase + nl;
        float w1v = 0.0f, w2v = 0.0f;
        if (n < OUT_CH) {
            w1v = W1[(size_t)k * OUT_CH + n];
            w2v = W2[(size_t)k * OUT_CH + n];
        }
        lw[nl * LW_STRIDE + k]          = w1v;
        lw[LW_MAT + nl * LW_STRIDE + k] = w2v;
    }
    __syncthreads();

    const int mr   = lane & 15;   // A: row in tile | B/C/D: col in tile
    const int hi   = lane >> 4;   // half-wave select
    const int koff = hi * 2;      // K sub-offset for A/B fragments

    v8f acc1 = {};
    v8f acc2 = {};

    // A: row (rowBase+mr), two consecutive K values starting at k0+koff
    const float* __restrict__ xrow = x + (size_t)(rowBase + mr) * IN_CH + koff;
    const float* bs1 = &lw[mr * LW_STRIDE + koff];
    const float* bs2 = bs1 + LW_MAT;

#pragma unroll 4
    for (int k0 = 0; k0 < IN_CH; k0 += 4) {
        v2f a   = *(const v2f*)(xrow + k0);   // global_load_b64
        v2f b1f = *(const v2f*)(bs1 + k0);    // ds_load_b64, conflict-free
        v2f b2f = *(const v2f*)(bs2 + k0);
        // D = A*B + C  (fp32 WMMA; EXEC is full here)
        acc1 = __builtin_amdgcn_wmma_f32_16x16x4_f32(
            false, a, false, b1f, (short)0, acc1, false, false);
        acc2 = __builtin_amdgcn_wmma_f32_16x16x4_f32(
            false, a, false, b2f, (short)0, acc2, false, false);
    }

    // C/D layout: VGPR r -> row r + hi*8, lane&15 -> column (coalesced stores)
    const int n = colBase + mr;
    if (n < OUT_CH) {
#pragma unroll
        for (int r = 0; r < 8; ++r) {
            int row = rowBase + r + hi * 8;
            xw1[(size_t)row * OUT_CH + n] = acc1[r];
            xw2[(size_t)row * OUT_CH + n] = acc2[r];
        }
    }
}

// ---------------------------------------------------------------------------
// 5) out = xw * dinv^2 (self-loop term) + bias. Also initializes d_out.
__global__ void self_bias_kernel(const float* __restrict__ xw1,
                                 const float* __restrict__ xw2,
                                 const float* __restrict__ dinv,
                                 const float* __restrict__ b1,
                                 const float* __restrict__ b2,
                                 float* __restrict__ mu,
                                 float* __restrict__ logstd) {
    int i = blockIdx.x * blockDim.x + threadIdx.x;
    if (i >= N_NODES * OUT_CH) return;
    int row = i / OUT_CH;
    int c   = i - row * OUT_CH;
    float w = dinv[row];
    w = w * w;
    mu[i]     = xw1[i] * w + b1[c];
    logstd[i] = xw2[i] * w + b2[c];
}

// ---------------------------------------------------------------------------
// 6) edge scatter: 50 threads per edge, 4 channels (float4) per thread.
// Consecutive threads share an edge row -> coalesced gathers; atomic f32
// scatter-adds land in L2 (xw + out are L2-resident on MI455X: 160 MB of
// 192 MB L2).
__global__ void scatter_kernel(const long long* __restrict__ src,
                               const long long* __restrict__ dst,
                               const float* __restrict__ dinv,
                               const float* __restrict__ xw1,
                               const float* __restrict__ xw2,
                               float* __restrict__ mu,
                               float* __restrict__ logstd,
                               int E) {
    int idx = blockIdx.x * blockDim.x + threadIdx.x;
    int total = E * 50;
    if (idx >= total) return;
    int e  = idx / 50;
    int c4 = (idx - e * 50) * 4;

    int s = (int)src[e];
    int d = (int)dst[e];
    float w = dinv[s] * dinv[d];

    const float4 v1 = *(const float4*)(xw1 + (size_t)s * OUT_CH + c4);
    const float4 v2 = *(const float4*)(xw2 + (size_t)s * OUT_CH + c4);

    float* pm = mu     + (size_t)d * OUT_CH + c4;
    float* pl = logstd + (size_t)d * OUT_CH + c4;
    atomicAdd(pm + 0, v1.x * w);
    atomicAdd(pm + 1, v1.y * w);
    atomicAdd(pm + 2, v1.z * w);
    atomicAdd(pm + 3, v1.w * w);
    atomicAdd(pl + 0, v2.x * w);
    atomicAdd(pl + 1, v2.y * w);
    atomicAdd(pl + 2, v2.z * w);
    atomicAdd(pl + 3, v2.w * w);
}

// ---------------------------------------------------------------------------
extern "C" void kernel_launch(void* const* d_in, const int* in_sizes, int n_in,
                              void* d_out, int out_size, void* d_ws, size_t ws_size,
                              hipStream_t stream) {
    const float*     x  = (const float*)d_in[0];
    const long long* ei = (const long long*)d_in[1];
    const float*     W1 = (const float*)d_in[2];
    const float*     b1 = (const float*)d_in[3];
    const float*     W2 = (const float*)d_in[4];
    const float*     b2 = (const float*)d_in[5];

    const int E = in_sizes[1] / 2;
    const long long* src = ei;
    const long long* dst = ei + E;

    float* mu     = (float*)d_out;
    float* logstd = (float*)d_out + (size_t)N_NODES * OUT_CH;

    // workspace layout: dinv | xw1 | xw2
    float* dinv = (float*)d_ws;
    size_t off  = ((size_t)N_NODES + 63) & ~(size_t)63;  // 256B-aligned
    float* xw1  = (float*)d_ws + off;
    float* xw2  = xw1 + (size_t)N_NODES * OUT_CH;

    // degree / normalization
    deg_init_kernel<<<(N_NODES + 255) / 256, 256, 0, stream>>>(dinv);
    deg_accum_kernel<<<(E + 255) / 256, 256, 0, stream>>>(dst, dinv, E);
    deg_rsqrt_kernel<<<(N_NODES + 255) / 256, 256, 0, stream>>>(dinv);

    // fused dual GEMM via fp32 WMMA (LDS-staged W slabs)
    dim3 ggrid((OUT_CH + 15) / 16, N_NODES / (16 * 5));  // (13, 625)
    gcn_gemm_wmma<<<ggrid, 160, 0, stream>>>(x, W1, W2, xw1, xw2);

    // self-loop + bias (initializes outputs)
    int nelem = N_NODES * OUT_CH;
    self_bias_kernel<<<(nelem + 255) / 256, 256, 0, stream>>>(
        xw1, xw2, dinv, b1, b2, mu, logstd);

    // edge scatter-add
    long long total = (long long)E * 50;
    scatter_kernel<<<(unsigned)((total + 255) / 256), 256, 0, stream>>>(
        src, dst, dinv, xw1, xw2, mu, logstd, E);
}